// GCN_85590108274951
// MI455X (gfx1250) — compile-verified
//
#include <hip/hip_runtime.h>
#include <math.h>

typedef __attribute__((ext_vector_type(2))) float v2f;
typedef __attribute__((ext_vector_type(8))) float v8f;

#define WPB 4  // waves per block in the GEMM

// ---------------------------------------------------------------------------
// Degree / normalization
// ---------------------------------------------------------------------------
__global__ void deg_init_k(float* __restrict__ deg, int n) {
    int i = blockIdx.x * blockDim.x + threadIdx.x;
    if (i < n) deg[i] = 1.0f;  // self loop
}

__global__ void deg_edges_k(float* __restrict__ deg, const int* __restrict__ dst, int e) {
    int i = blockIdx.x * blockDim.x + threadIdx.x;
    if (i < e) atomicAdd(&deg[dst[i]], 1.0f);
}

__global__ void dinv_k(float* __restrict__ deg, int n) {
    int i = blockIdx.x * blockDim.x + threadIdx.x;
    if (i < n) deg[i] = rsqrtf(fmaxf(deg[i], 1.0f));  // in place: deg -> dinv
}

// ---------------------------------------------------------------------------
// fp32 WMMA GEMM: Out[M,NOUT] = A[M,K] @ W[K,NOUT]   (M % 16 == 0)
// One wave computes one 16x16 tile via V_WMMA_F32_16X16X4_F32.
// K and NOUT are compile-time so every load/store offset is an immediate;
// B fragments preloaded into K/4 v2f registers -> hot loop is b64-load + wmma.
// ---------------------------------------------------------------------------
template <int K, int NOUT>
__global__ __launch_bounds__(32 * WPB)
void gemm_wmma_f32(const float* __restrict__ A, const float* __restrict__ W,
                   float* __restrict__ Out, int M) {
    const int lane = threadIdx.x & 31;
    const int wave = threadIdx.x >> 5;
    const int mt   = blockIdx.x * WPB + wave;
    if (mt * 16 >= M) return;                 // whole-wave uniform exit; EXEC stays all-1
    const int nt  = blockIdx.y;
    const int r16 = lane & 15;
    const int kh  = (lane >> 4) << 1;         // 0 for lanes 0-15, 2 for lanes 16-31
    const int col = nt * 16 + r16;            // B/D column held by this lane

    // Preload loop-invariant B fragments: lane holds W[k0+kh][col], W[k0+kh+1][col]
    const float* wbase = W + (size_t)kh * NOUT + col;   // per-lane base; k0 offsets constant
    v2f bfrag[K / 4];
#pragma unroll
    for (int k0 = 0; k0 < K; k0 += 4) {
        bfrag[k0 / 4].x = wbase[(size_t)k0 * NOUT];
        bfrag[k0 / 4].y = wbase[(size_t)k0 * NOUT + NOUT];
    }

    // A 16x4 fragments: lane holds A[row][k0+kh .. k0+kh+1] (8B aligned: K even, kh even)
    const float* arow = A + (size_t)(mt * 16 + r16) * K + kh;

    v8f acc = {};
#pragma unroll
    for (int k0 = 0; k0 < K; k0 += 4) {
        v2f a = *(const v2f*)(arow + k0);
        acc = __builtin_amdgcn_wmma_f32_16x16x4_f32(
            /*neg_a=*/false, a, /*neg_b=*/false, bfrag[k0 / 4],
            /*c_mod=*/(short)0, acc, /*reuse_a=*/false, /*reuse_b=*/false);
    }

    // C/D layout: VGPR r -> M = r (lanes 0-15) / M = r+8 (lanes 16-31), N = lane&15
    const int mrow = mt * 16 + ((lane >> 4) << 3);
    float* obase = Out + (size_t)mrow * NOUT + col;     // r offsets are immediates
#pragma unroll
    for (int r = 0; r < 8; ++r)
        obase[(size_t)r * NOUT] = acc[r];
}

// ---------------------------------------------------------------------------
// GCN aggregation: out = selfloop + bias, then edge atomics, then SiLU
// ---------------------------------------------------------------------------
template <int F>
__global__ void agg_init_k(const float* __restrict__ h, float* __restrict__ out,
                           const float* __restrict__ dinv, const float* __restrict__ bias,
                           int n) {
    int idx = blockIdx.x * blockDim.x + threadIdx.x;
    if (idx >= n * F) return;
    int i = idx / F;            // constant divisor -> mul/shift
    int f = idx - i * F;
    float di = dinv[i];
    out[idx] = h[idx] * di * di + bias[f];
}

template <int F4>
__global__ void agg_edges_k(const float* __restrict__ h, float* __restrict__ out,
                            const int* __restrict__ src, const int* __restrict__ dst,
                            const float* __restrict__ dinv, int e) {
    int idx = blockIdx.x * blockDim.x + threadIdx.x;
    if (idx >= e * F4) return;
    int ei = idx / F4;          // constant divisor -> mul/shift
    int j  = idx - ei * F4;
    int s = src[ei];
    int d = dst[ei];
    float norm = dinv[s] * dinv[d];
    const float4 hv = ((const float4*)h)[(size_t)s * F4 + j];  // L2-resident gather
    float* o = out + ((size_t)d * F4 + j) * 4;
    atomicAdd(o + 0, hv.x * norm);
    atomicAdd(o + 1, hv.y * norm);
    atomicAdd(o + 2, hv.z * norm);
    atomicAdd(o + 3, hv.w * norm);
}

__global__ void silu_k(float* __restrict__ x, int n) {
    int i = blockIdx.x * blockDim.x + threadIdx.x;
    if (i < n) {
        float v = x[i];
        x[i] = v / (1.0f + __expf(-v));
    }
}

// ---------------------------------------------------------------------------
// Pooling + log_softmax
// ---------------------------------------------------------------------------
__device__ __forceinline__ void atomicMaxFloat(float* addr, float val) {
    if (val >= 0.0f) atomicMax((int*)addr, __float_as_int(val));
    else             atomicMin((unsigned int*)addr, __float_as_uint(val));
}

__global__ void pool_init_k(float* __restrict__ gmax, float* __restrict__ gsum,
                            float* __restrict__ gcnt, int G) {
    int i = blockIdx.x * blockDim.x + threadIdx.x;
    if (i < G * 32) {
        gmax[i] = -INFINITY;
        gsum[i] = 0.0f;
        if (i < G) gcnt[i] = 0.0f;
    }
}

__global__ void pool_scatter_k(const float* __restrict__ h, const int* __restrict__ batch,
                               float* __restrict__ gmax, float* __restrict__ gsum,
                               float* __restrict__ gcnt, int n) {
    int idx = blockIdx.x * blockDim.x + threadIdx.x;
    if (idx >= n * 32) return;
    int i = idx >> 5;
    int c = idx & 31;
    int g = batch[i];
    float v = h[idx];
    atomicMaxFloat(&gmax[g * 32 + c], v);
    atomicAdd(&gsum[g * 32 + c], v);
    if (c == 0) atomicAdd(&gcnt[g], 1.0f);
}

__global__ void final_k(const float* __restrict__ gmax, const float* __restrict__ gsum,
                        const float* __restrict__ gcnt, float* __restrict__ out) {
    const int g = blockIdx.x;   // one block (64 threads) per graph
    const int t = threadIdx.x;  // 0..63: [0,32)=gmax, [32,64)=gmean
    float v = (t < 32) ? gmax[g * 32 + t]
                       : gsum[g * 32 + (t - 32)] / fmaxf(gcnt[g], 1.0f);
    __shared__ float sm[64];
    sm[t] = v;
    __syncthreads();
    for (int s = 32; s > 0; s >>= 1) { if (t < s) sm[t] = fmaxf(sm[t], sm[t + s]); __syncthreads(); }
    float mx = sm[0];
    __syncthreads();
    sm[t] = __expf(v - mx);
    __syncthreads();
    for (int s = 32; s > 0; s >>= 1) { if (t < s) sm[t] += sm[t + s]; __syncthreads(); }
    float lse = mx + logf(sm[0]);
    out[g * 64 + t] = v - lse;
}

// ---------------------------------------------------------------------------
// Launch
// ---------------------------------------------------------------------------
static inline unsigned cdiv(long a, long b) { return (unsigned)((a + b - 1) / b); }

template <int F>
static void run_layer(const float* in, const float* W, const float* bias,
                      float* tmp, float* out, const float* dinv,
                      const int* srcI, const int* dstI,
                      int Nn, int E, bool act, hipStream_t stream) {
    const int Mt = (Nn + 15) / 16;
    dim3 gg(cdiv(Mt, WPB), F / 16);
    gemm_wmma_f32<96, F><<<gg, 32 * WPB, 0, stream>>>(in, W, tmp, Nn);
    const long tot = (long)Nn * F;
    agg_init_k<F><<<cdiv(tot, 256), 256, 0, stream>>>(tmp, out, dinv, bias, Nn);
    const long etot = (long)E * (F / 4);
    agg_edges_k<F / 4><<<cdiv(etot, 256), 256, 0, stream>>>(tmp, out, srcI, dstI, dinv, E);
    if (act) silu_k<<<cdiv(tot, 256), 256, 0, stream>>>(out, (int)tot);
}

extern "C" void kernel_launch(void* const* d_in, const int* in_sizes, int n_in,
                              void* d_out, int out_size, void* d_ws, size_t ws_size,
                              hipStream_t stream) {
    const float* x   = (const float*)d_in[0];
    const float* W0  = (const float*)d_in[1];
    const float* b0  = (const float*)d_in[2];
    const float* W1  = (const float*)d_in[3];
    const float* b1  = (const float*)d_in[4];
    const float* W2  = (const float*)d_in[5];
    const float* b2  = (const float*)d_in[6];
    const int*   eix = (const int*)d_in[7];
    const int*   bat = (const int*)d_in[8];

    const int Nn = in_sizes[0] / 96;   // 100000
    const int E  = in_sizes[7] / 2;    // 800000
    const int G  = out_size / 64;      // 128
    const int* srcI = eix;             // edge_index[0]
    const int* dstI = eix + E;         // edge_index[1]

    float* ws   = (float*)d_ws;
    float* dinv = ws;                              // N
    float* buf0 = dinv + Nn;                       // N*96 (GEMM output / gather source)
    float* buf1 = buf0 + (size_t)Nn * 96;          // N*96 (aggregated / activation)
    float* gmax = buf1 + (size_t)Nn * 96;          // G*32
    float* gsum = gmax + (size_t)G * 32;           // G*32
    float* gcnt = gsum + (size_t)G * 32;           // G

    // degree -> dinv (computed once; identical across layers)
    deg_init_k <<<cdiv(Nn, 256), 256, 0, stream>>>(dinv, Nn);
    deg_edges_k<<<cdiv(E, 256),  256, 0, stream>>>(dinv, dstI, E);
    dinv_k     <<<cdiv(Nn, 256), 256, 0, stream>>>(dinv, Nn);

    // layer 0: x(96) -> buf1(96), SiLU
    run_layer<96>(x,    W0, b0, buf0, buf1, dinv, srcI, dstI, Nn, E, true,  stream);
    // layer 1: buf1(96) -> buf1(96), SiLU
    run_layer<96>(buf1, W1, b1, buf0, buf1, dinv, srcI, dstI, Nn, E, true,  stream);
    // layer 2: buf1(96) -> buf1(32), no activation
    run_layer<32>(buf1, W2, b2, buf0, buf1, dinv, srcI, dstI, Nn, E, false, stream);

    // global max + mean pool, then log_softmax
    pool_init_k   <<<cdiv((long)G * 32, 256), 256, 0, stream>>>(gmax, gsum, gcnt, G);
    pool_scatter_k<<<cdiv((long)Nn * 32, 256), 256, 0, stream>>>(buf1, bat, gmax, gsum, gcnt, Nn);
    final_k       <<<G, 64, 0, stream>>>(gmax, gsum, gcnt, (float*)d_out);

    (void)n_in; (void)ws_size; (void)in_sizes;
}